// knn_matrix_7937099563612
// MI455X (gfx1250) — compile-verified
//
#include <hip/hip_runtime.h>
#include <hip/hip_bf16.h>

// Problem constants (fixed by the reference).
constexpr int kB = 8;
constexpr int kN = 4096;
constexpr int kD = 64;
constexpr int kK = 16;
constexpr int kTiles = kN / 16;   // 256 column tiles per row

typedef __attribute__((ext_vector_type(2))) float v2f;
typedef __attribute__((ext_vector_type(8))) float v8f;

// ---------------------------------------------------------------------------
// Kernel 1: per-point squared norms  ||x_i||^2  -> workspace
// ---------------------------------------------------------------------------
__global__ void knn_rowsq_kernel(const float* __restrict__ x,
                                 float* __restrict__ sq) {
    int r = blockIdx.x * blockDim.x + threadIdx.x;
    if (r >= kB * kN) return;
    const float4* p = (const float4*)(x + (size_t)r * kD);
    float s = 0.0f;
#pragma unroll
    for (int i = 0; i < kD / 4; ++i) {
        float4 v = p[i];
        s += v.x * v.x + v.y * v.y + v.z * v.z + v.w * v.w;
    }
    sq[r] = s;
}

// ---------------------------------------------------------------------------
// Kernel 2: fused WMMA Gram-matrix + per-row top-K (neg squared distance).
// Block = 128 threads = 4 waves. Each wave owns 32 rows (two 16-row WMMA
// A-blocks) and sweeps all 256 column tiles of its batch.
// ---------------------------------------------------------------------------
__launch_bounds__(128, 1)
__global__ void knn_topk_kernel(const float* __restrict__ x,
                                const float* __restrict__ xsq,
                                float* __restrict__ out) {
    // Per-wave private LDS slab: 32 rows x 16 cols, padded stride 17 (no bank
    // conflicts: 17 coprime with 64 banks).
    __shared__ float lds[4][32 * 17];

    const int wave = threadIdx.x >> 5;      // 0..3 (wave32)
    const int lane = threadIdx.x & 31;      // 0..31
    const int half = lane >> 4;             // 0 or 1
    const int l16  = lane & 15;

    const int b       = blockIdx.x >> 5;    // 8 batches * 32 row-tiles = 256 blocks
    const int rowTile = blockIdx.x & 31;    // 128 rows per block
    const int rb0     = rowTile * 128 + wave * 32;   // first row (in-batch) of this wave

    const float* xb  = x   + (size_t)b * kN * kD;
    const float* sqb = xsq + (size_t)b * kN;

    // --- Preload A fragments for both 16-row blocks (kept live all loop) ---
    // A 16x4 f32 layout: lane L holds row L%16; VGPR v holds K = (L/16)*2 + v.
    v2f a0[16], a1[16];
    const float* arow0 = xb + (size_t)(rb0 + l16) * kD;
    const float* arow1 = xb + (size_t)(rb0 + 16 + l16) * kD;
#pragma unroll
    for (int ks = 0; ks < 16; ++ks) {
        const int k = ks * 4 + half * 2;           // even -> 8B aligned
        a0[ks] = *(const v2f*)(arow0 + k);
        a1[ks] = *(const v2f*)(arow1 + k);
    }

    // Selection: lane L owns in-batch row (rb0 + L).
    const float sqSel = sqb[rb0 + lane];

    float dlist[kK];   // neg_adj values, sorted descending (best first)
    int   ilist[kK];   // in-batch column indices
#pragma unroll
    for (int j = 0; j < kK; ++j) { dlist[j] = -3.0e38f; ilist[j] = 0; }

    float* myLds = lds[wave];

    for (int ct = 0; ct < kTiles; ++ct) {
        const int colbase = ct * 16;

        // --- Gram tile via fp32 WMMA: C = A(16x64) * B(64x16), K-step 4 ---
        const float* brow = xb + (size_t)(colbase + l16) * kD;
        v8f c0 = {};
        v8f c1 = {};
#pragma unroll
        for (int ks = 0; ks < 16; ++ks) {
            const int k = ks * 4 + half * 2;
            v2f bf = *(const v2f*)(brow + k);
            c0 = __builtin_amdgcn_wmma_f32_16x16x4_f32(
                     false, a0[ks], false, bf, (short)0, c0, false, false);
            c1 = __builtin_amdgcn_wmma_f32_16x16x4_f32(
                     false, a1[ks], false, bf, (short)0, c1, false, false);
        }

        // Fold in -||x_col||^2 while scattering C to LDS.
        // C layout: VGPR v, lane L -> M = v + 8*(L/16), N = L%16.
        const float csq = sqb[colbase + l16];
#pragma unroll
        for (int v = 0; v < 8; ++v) {
            const int r0 = v + 8 * half;
            myLds[r0 * 17 + l16]        = 2.0f * c0[v] - csq;   // rows rb0+0..15
            myLds[(16 + r0) * 17 + l16] = 2.0f * c1[v] - csq;   // rows rb0+16..31
        }
        // Wave-private LDS slab: only need same-wave DS completion, no barrier.
        asm volatile("s_wait_dscnt 0x0" ::: "memory");

        // --- Threshold-gated sorted insert: lane owns one full row ---
        const float* rowp = myLds + lane * 17;
#pragma unroll
        for (int j = 0; j < 16; ++j) {
            const float cand = rowp[j] - sqSel;    // neg_adj = 2*dot - |r|^2 - |c|^2
            if (cand > dlist[kK - 1]) {
                dlist[kK - 1] = cand;
                ilist[kK - 1] = colbase + j;
                // One bubble pass restores descending order.
#pragma unroll
                for (int s = kK - 1; s > 0; --s) {
                    if (dlist[s] > dlist[s - 1]) {
                        float td = dlist[s]; dlist[s] = dlist[s - 1]; dlist[s - 1] = td;
                        int   ti = ilist[s]; ilist[s] = ilist[s - 1]; ilist[s - 1] = ti;
                    }
                }
            }
        }
        asm volatile("s_wait_dscnt 0x0" ::: "memory");
    }

    // --- Emit: edge_index[0] (=nn idx), edge_index[1] (=center), val ---
    const int   n = rb0 + lane;                    // in-batch row
    const long  g = (long)b * kN + n;              // flat node id
    const size_t BNK = (size_t)kB * kN * kK;
    float* nnOut = out;
    float* ctOut = out + BNK;
    float* vOut  = out + 2 * BNK;
#pragma unroll
    for (int j = 0; j < kK; ++j) {
        nnOut[(size_t)g * kK + j] = (float)(b * kN + ilist[j]);  // exact (<2^24)
        ctOut[(size_t)g * kK + j] = (float)g;
        vOut[(size_t)g * kK + j]  = dlist[j];
    }
}

// ---------------------------------------------------------------------------
extern "C" void kernel_launch(void* const* d_in, const int* in_sizes, int n_in,
                              void* d_out, int out_size, void* d_ws, size_t ws_size,
                              hipStream_t stream) {
    (void)in_sizes; (void)n_in; (void)out_size; (void)ws_size;
    const float* x   = (const float*)d_in[0];   // (B*N, D) fp32
    float*       xsq = (float*)d_ws;            // B*N fp32 squared norms
    float*       out = (float*)d_out;           // [nn_idx | center_idx | val]

    knn_rowsq_kernel<<<(kB * kN + 255) / 256, 256, 0, stream>>>(x, xsq);
    knn_topk_kernel<<<kB * (kN / 128), 128, 0, stream>>>(x, xsq, out);
}